// EGraphSAGE_62723702391486
// MI455X (gfx1250) — compile-verified
//
#include <hip/hip_runtime.h>
#include <hip/hip_bf16.h>

// ---------------- problem constants ----------------
constexpr int N_NODES = 50000;
constexpr int N_EDGES = 800000;
constexpr int D_FEAT  = 64;
constexpr int H       = 128;

typedef __attribute__((ext_vector_type(2))) float v2f;
typedef __attribute__((ext_vector_type(8))) float v8f;

__device__ __forceinline__ void fadd(float* p, float v) {
    // lowers to global_atomic_add_f32 (L2 RMW, relaxed, device scope)
    __hip_atomic_fetch_add(p, v, __ATOMIC_RELAXED, __HIP_MEMORY_SCOPE_AGENT);
}

// ---------------- kernel 0: zero scatter accumulators ----------------
__global__ __launch_bounds__(256) void k_zero(float* p, int n) {
    int i = blockIdx.x * 256 + threadIdx.x;
    int stride = gridDim.x * 256;
    for (; i < n; i += stride) p[i] = 0.0f;
}

// ---------------- kernel 1: fused dual scatter-sum + counts ----------------
// one thread per (edge, 4-feature chunk): b128 read, 8 f32 atomics
__global__ __launch_bounds__(256) void k_scatter(const float* __restrict__ ea,
                                                 const int* __restrict__ eidx,
                                                 float* __restrict__ s_dst,
                                                 float* __restrict__ s_src,
                                                 float* __restrict__ c_dst,
                                                 float* __restrict__ c_src) {
    int idx = blockIdx.x * 256 + threadIdx.x;          // < 12.8M
    if (idx >= N_EDGES * (D_FEAT / 4)) return;
    int e  = idx >> 4;
    int f4 = (idx & 15) << 2;
    float4 v = *(const float4*)(ea + e * D_FEAT + f4);
    int s = eidx[e];
    int d = eidx[N_EDGES + e];
    float* ps = s_src + s * D_FEAT + f4;
    float* pd = s_dst + d * D_FEAT + f4;
    fadd(ps + 0, v.x); fadd(ps + 1, v.y); fadd(ps + 2, v.z); fadd(ps + 3, v.w);
    fadd(pd + 0, v.x); fadd(pd + 1, v.y); fadd(pd + 2, v.z); fadd(pd + 3, v.w);
    if (f4 == 0) {
        fadd(&c_src[s], 1.0f);
        fadd(&c_dst[d], 1.0f);
    }
}

// ---------------- kernel 2: fused node MLP (two layers) via f32 WMMA ----------
// 256 threads (8 waves) per block; block handles 128 nodes.
// LDS layout (dynamic, floats):
//   X   [128][260]  : cols [0,64)=mean_dst, [64,128)=mean_src, [128,256)=h0
//   W0t [128][132]  : W0 transposed, W0t[col][k] = W0[k][col]
//   W1t [128][196]  : W1 transposed, W1t[col][k] = W1[k][col]
// pad strides (260/132/196 ≡ 4 mod 64) -> conflict-free column reads
#define XSTR  260
#define W0STR 132
#define W1STR 196
#define X_OFF  0
#define W0_OFF (128 * XSTR)               // 33280
#define W1_OFF (W0_OFF + 128 * W0STR)     // 50176
#define LDS_FLOATS (W1_OFF + 128 * W1STR) // 75264 -> 301,056 bytes

__device__ __forceinline__ float sigmoidf(float x) {
    return 1.0f / (1.0f + __expf(-x));
}

__global__ __launch_bounds__(256) void k_mlp(const float* __restrict__ s_dst,
                                             const float* __restrict__ s_src,
                                             const float* __restrict__ c_dst,
                                             const float* __restrict__ c_src,
                                             const float* __restrict__ W0,
                                             const float* __restrict__ b0,
                                             const float* __restrict__ W1,
                                             const float* __restrict__ b1,
                                             float* __restrict__ h1out) {
    extern __shared__ float L[];
    float* X   = L + X_OFF;
    float* W0t = L + W0_OFF;
    float* W1t = L + W1_OFF;

    const int tid  = threadIdx.x;
    const int lane = tid & 31;
    const int wave = tid >> 5;
    const int nb   = blockIdx.x * 128;

    // ---- phase 0: stage transposed weights (coalesced global reads) ----
    for (int i = tid; i < H * H; i += 256) {          // W0: [128][128]
        int k = i >> 7, c = i & 127;
        W0t[c * W0STR + k] = W0[k * H + c];
    }
    for (int i = tid; i < (H + D_FEAT) * H; i += 256) { // W1: [192][128]
        int k = i >> 7, c = i & 127;
        W1t[c * W1STR + k] = W1[k * H + c];
    }

    // ---- phase 1: stage means into LDS (cooperative) ----
    for (int i = tid; i < 128 * 128; i += 256) {
        int r = i >> 7, c = i & 127;
        int node = nb + r;
        float v = 0.0f;
        if (node < N_NODES) {
            if (c < 64) v = s_dst[node * D_FEAT + c]        / fmaxf(c_dst[node], 1.0f);
            else        v = s_src[node * D_FEAT + (c - 64)] / fmaxf(c_src[node], 1.0f);
        }
        X[r * XSTR + c] = v;
    }
    __syncthreads();

    // WMMA lane mapping (16x16x4 f32):
    //  A: lane&15 = M row, VGPR pair = K+kh, K+kh+1 (kh = 0 for lanes 0-15, 2 for 16-31)
    //  B: lane&15 = N col, same K split
    //  D: VGPR v holds (row = v + 8*(lane>=16), col = lane&15)
    const int m     = lane & 15;
    const int khalf = (lane >> 4) << 1;
    const int rowA  = wave * 16 + m;
    const int rbase = wave * 16 + ((lane >> 4) << 3);

    // ---- phase 2: layer 0, K = 128 ----
    for (int ct = 0; ct < 8; ++ct) {
        v8f acc = {};
        const float* wcol = W0t + (ct * 16 + m) * W0STR;
        for (int k = 0; k < 2 * D_FEAT; k += 4) {
            float2 av = *(const float2*)(X + rowA * XSTR + k + khalf);   // ds_load_b64
            float2 bv = *(const float2*)(wcol + k + khalf);              // ds_load_b64
            v2f a, b;
            a.x = av.x; a.y = av.y;
            b.x = bv.x; b.y = bv.y;
            acc = __builtin_amdgcn_wmma_f32_16x16x4_f32(false, a, false, b,
                                                        (short)0, acc, false, false);
        }
        float bias = b0[ct * 16 + m];
#pragma unroll
        for (int v = 0; v < 8; ++v) {
            float h = sigmoidf(acc[v] + bias);
            X[(rbase + v) * XSTR + 128 + ct * 16 + m] = h;   // wave-private rows
        }
    }
    __syncthreads();

    // ---- phase 3: layer 1, K = 192  (W1 rows 0..127 <- h0, rows 128..191 <- agg)
    for (int ct = 0; ct < 8; ++ct) {
        v8f acc = {};
        const float* wcol = W1t + (ct * 16 + m) * W1STR;
        for (int k = 0; k < H + D_FEAT; k += 4) {
            int kk  = k + khalf;                         // even; pair never crosses 128
            int xc0 = (kk < H) ? (128 + kk) : (kk - 64);
            float2 av = *(const float2*)(X + rowA * XSTR + xc0);   // ds_load_b64
            float2 bv = *(const float2*)(wcol + kk);               // ds_load_b64
            v2f a, b;
            a.x = av.x; a.y = av.y;
            b.x = bv.x; b.y = bv.y;
            acc = __builtin_amdgcn_wmma_f32_16x16x4_f32(false, a, false, b,
                                                        (short)0, acc, false, false);
        }
        float bias = b1[ct * 16 + m];
#pragma unroll
        for (int v = 0; v < 8; ++v) {
            int node = nb + rbase + v;
            if (node < N_NODES) {
                h1out[node * H + ct * 16 + m] = sigmoidf(acc[v] + bias);
            }
        }
    }
}

// ---------------- kernel 3: edge classifier + embedding stream-out -----------
// one wave per edge; lane l owns features [4l,4l+4) of each endpoint.
__global__ __launch_bounds__(256) void k_edge(const int* __restrict__ eidx,
                                              const float* __restrict__ h1,
                                              const float* __restrict__ Wf,
                                              const float* __restrict__ bf,
                                              float* __restrict__ logits,
                                              float* __restrict__ emb) {
    int lane = threadIdx.x & 31;
    int e = blockIdx.x * 8 + (threadIdx.x >> 5);
    if (e >= N_EDGES) return;
    int s = eidx[e];
    int d = eidx[N_EDGES + e];

    float4 hs = *(const float4*)(h1 + s * H + lane * 4);
    float4 hd = *(const float4*)(h1 + d * H + lane * 4);
    float4 ws = *(const float4*)(Wf + lane * 4);
    float4 wd = *(const float4*)(Wf + H + lane * 4);

    float4* eb = (float4*)(emb + (size_t)e * 2 * H);
    eb[lane]      = hs;
    eb[32 + lane] = hd;

    float p = hs.x * ws.x + hs.y * ws.y + hs.z * ws.z + hs.w * ws.w
            + hd.x * wd.x + hd.y * wd.y + hd.z * wd.z + hd.w * wd.w;
#pragma unroll
    for (int off = 16; off > 0; off >>= 1) p += __shfl_xor(p, off, 32);
    if (lane == 0) logits[e] = p + bf[0];
}

// ---------------- launch ----------------
extern "C" void kernel_launch(void* const* d_in, const int* in_sizes, int n_in,
                              void* d_out, int out_size, void* d_ws, size_t ws_size,
                              hipStream_t stream) {
    const float* edge_attr = (const float*)d_in[0];
    const int*   edge_idx  = (const int*)  d_in[1];
    const float* W0        = (const float*)d_in[2];
    const float* b0        = (const float*)d_in[3];
    const float* W1        = (const float*)d_in[4];
    const float* b1        = (const float*)d_in[5];
    const float* Wf        = (const float*)d_in[6];
    const float* bf        = (const float*)d_in[7];

    float* ws    = (float*)d_ws;
    float* s_dst = ws;                                   // N * 64
    float* s_src = s_dst + N_NODES * D_FEAT;             // N * 64
    float* c_dst = s_src + N_NODES * D_FEAT;             // N
    float* c_src = c_dst + N_NODES;                      // N
    float* h1    = c_src + N_NODES;                      // N * 128

    float* logits = (float*)d_out;                       // E
    float* emb    = logits + N_EDGES;                    // E * 256

    int nzero = 2 * N_NODES * D_FEAT + 2 * N_NODES;
    k_zero<<<4096, 256, 0, stream>>>(ws, nzero);

    int scatter_blocks = (N_EDGES * (D_FEAT / 4) + 255) / 256;
    k_scatter<<<scatter_blocks, 256, 0, stream>>>(edge_attr, edge_idx,
                                                  s_dst, s_src, c_dst, c_src);

    int mlp_blocks = (N_NODES + 127) / 128;
    size_t lds_bytes = (size_t)LDS_FLOATS * sizeof(float);   // 301,056 B
    k_mlp<<<mlp_blocks, 256, lds_bytes, stream>>>(s_dst, s_src, c_dst, c_src,
                                                  W0, b0, W1, b1, h1);

    int edge_blocks = (N_EDGES + 7) / 8;
    k_edge<<<edge_blocks, 256, 0, stream>>>(edge_idx, h1, Wf, bf, logits, emb);
}